// MPNetm_19267223290692
// MI455X (gfx1250) — compile-verified
//
#include <hip/hip_runtime.h>
#include <hip/hip_bf16.h>
#include <math.h>

#define N_NODES 20000
#define N_EDGES 500000
#define D       256
#define NUM_REL 5
#define LL_OUT  16

// All GEMMs in this net have K == 256 (fc1 is split into two K=256 passes).
#define KDIM    256
#define BT_PAD  260   // LDS row stride (floats): 8B-aligned pairs, banks (4m+k)%64 conflict-free

typedef float v2f __attribute__((ext_vector_type(2)));
typedef float v8f __attribute__((ext_vector_type(8)));

#define GF_BIAS 1
#define GF_RELU 2
#define GF_ACC  4

// ---------------------------------------------------------------------------
// fp32 WMMA GEMM: C[nRows x Dout] = op( A[nRows x 256] @ B[256 x Dout] )
// Block = 256 threads = 8 waves; block tile = 128 rows x 32 cols.
// Each wave owns one 16-row strip and TWO adjacent 16x16 C tiles, sharing its
// A fragment between them. B panel staged transposed in LDS once per block:
//   per 4-K step: 1 global b64 (A) + 2 ds b64 (B) -> 2 v_wmma_f32_16x16x4_f32.
// ---------------------------------------------------------------------------
__global__ __launch_bounds__(256) void wmma_gemm_f32(
    const float* __restrict__ A, const float* __restrict__ B,
    const float* __restrict__ bias, float* __restrict__ C,
    int nRows, int Dout, int flags)
{
    __shared__ float Bt[32][BT_PAD];          // B^T panel: Bt[local_col][k]

    const int tid  = threadIdx.x;
    const int lane = tid & 31;
    const int wave = tid >> 5;
    const int m    = lane & 15;               // N within tile / A row within tile
    const int h    = lane >> 4;               // half-wave -> K pair selector
    const int colBase = blockIdx.x * 32;
    const int rowBase = blockIdx.y * 128 + wave * 16;

    // ---- stage B panel (256 x 32) into LDS, transposed ----
    {
        const int cl   = tid & 31;            // local col 0..31
        const int colg = colBase + cl;
        const bool ok  = colg < Dout;
        for (int k = tid >> 5; k < KDIM; k += 8)      // coalesced 128B rows
            Bt[cl][k] = ok ? B[(size_t)k * Dout + colg] : 0.0f;
    }
    __syncthreads();

    if (rowBase >= nRows) return;

    int ra = rowBase + m;
    if (ra >= nRows) ra = nRows - 1;          // clamp loads on partial tile
    const float* __restrict__ Arow = A + (size_t)ra * KDIM;

    // Warm the A row (8 x 128B cachelines) -> global_prefetch_b8
#pragma unroll
    for (int pf = 0; pf < KDIM * 4; pf += 128)
        __builtin_prefetch((const char*)Arow + pf, 0, 3);

    v8f c0 = {0.f, 0.f, 0.f, 0.f, 0.f, 0.f, 0.f, 0.f};
    v8f c1 = {0.f, 0.f, 0.f, 0.f, 0.f, 0.f, 0.f, 0.f};

    for (int k0 = 0; k0 < KDIM; k0 += 4) {
        // A 16x4 fragment: lane holds K = k0+2h, k0+2h+1 (aligned float2)
        v2f a  = *(const v2f*)(Arow + k0 + 2 * h);
        // B 4x16 fragments from LDS (aligned b64, bank-conflict-free)
        v2f b0 = *(const v2f*)&Bt[m][k0 + 2 * h];
        v2f b1 = *(const v2f*)&Bt[m + 16][k0 + 2 * h];
        c0 = __builtin_amdgcn_wmma_f32_16x16x4_f32(
                 false, a, false, b0, (short)0, c0, false, false);
        c1 = __builtin_amdgcn_wmma_f32_16x16x4_f32(
                 false, a, false, b1, (short)0, c1, false, false);
    }

    // ---- epilogue: bias / accumulate / relu, two column tiles ----
#pragma unroll
    for (int t = 0; t < 2; ++t) {
        const int col = colBase + 16 * t + m;
        if (col >= Dout) break;
        const float bv = (flags & GF_BIAS) ? bias[col] : 0.0f;
        const v8f& c = t ? c1 : c0;
#pragma unroll
        for (int v = 0; v < 8; ++v) {
            const int r = rowBase + v + 8 * h;    // C: M = v + 8h, N = m
            if (r < nRows) {
                const size_t idx = (size_t)r * Dout + col;
                float val = c[v] + bv;
                if (flags & GF_ACC)  val += C[idx];
                if (flags & GF_RELU) val = fmaxf(val, 0.0f);
                C[idx] = val;
            }
        }
    }
}

// ---------------------------------------------------------------------------
// Edge scatter: for edges of type==rel, agg[src] += hr[dst], cnt[src] += 1.
// 64 threads (2 waves) per edge, float4 per lane -> 256 floats/edge.
// ---------------------------------------------------------------------------
__global__ __launch_bounds__(256) void edge_scatter(
    const float* __restrict__ hr, const int* __restrict__ src,
    const int* __restrict__ dst, const int* __restrict__ etype,
    int rel, float* __restrict__ agg, float* __restrict__ cnt)
{
    const int e      = blockIdx.x * 4 + (threadIdx.x >> 6);
    const int lane64 = threadIdx.x & 63;
    if (e >= N_EDGES) return;
    if (etype[e] != rel) return;              // uniform across the 64-group
    const int s = src[e];
    const int t = dst[e];
    const float4 v = *(const float4*)(hr + (size_t)t * D + lane64 * 4);
    float* out = agg + (size_t)s * D + lane64 * 4;
    atomicAdd(out + 0, v.x);
    atomicAdd(out + 1, v.y);
    atomicAdd(out + 2, v.z);
    atomicAdd(out + 3, v.w);
    if (lane64 == 0) atomicAdd(cnt + s, 1.0f);
}

// ---------------------------------------------------------------------------
__global__ void zero_f32(float* __restrict__ p, size_t n)
{
    size_t i = (size_t)blockIdx.x * blockDim.x + threadIdx.x;
    const size_t stride = (size_t)gridDim.x * blockDim.x;
    for (; i < n; i += stride) p[i] = 0.0f;
}

// h_out = relu(agg / max(cnt,1) + base)
__global__ void conv_finalize(
    const float* __restrict__ agg, const float* __restrict__ cnt,
    const float* __restrict__ base, float* __restrict__ hout)
{
    const size_t i = (size_t)blockIdx.x * blockDim.x + threadIdx.x;
    if (i >= (size_t)N_NODES * D) return;
    const int row = (int)(i >> 8);            // D == 256
    const float rs = 1.0f / fmaxf(cnt[row], 1.0f);
    hout[i] = fmaxf(agg[i] * rs + base[i], 0.0f);
}

__global__ void log_softmax16(const float* __restrict__ logits,
                              float* __restrict__ out)
{
    const int n = blockIdx.x * blockDim.x + threadIdx.x;
    if (n >= N_NODES) return;
    const float* p = logits + (size_t)n * LL_OUT;
    float mx = p[0];
#pragma unroll
    for (int j = 1; j < LL_OUT; ++j) mx = fmaxf(mx, p[j]);
    float s = 0.0f;
#pragma unroll
    for (int j = 0; j < LL_OUT; ++j) s += __expf(p[j] - mx);
    const float lse = mx + __logf(s);
    float* o = out + (size_t)n * LL_OUT;
#pragma unroll
    for (int j = 0; j < LL_OUT; ++j) o[j] = p[j] - lse;
}

// ---------------------------------------------------------------------------
extern "C" void kernel_launch(void* const* d_in, const int* in_sizes, int n_in,
                              void* d_out, int out_size, void* d_ws, size_t ws_size,
                              hipStream_t stream)
{
    const float* x      = (const float*)d_in[0];
    const int*   eidx   = (const int*)  d_in[1];
    const int*   etype  = (const int*)  d_in[2];
    const float* rel_W  = (const float*)d_in[3];  // [4][5][256][256]
    const float* root_W = (const float*)d_in[4];  // [4][256][256]
    const float* biasv  = (const float*)d_in[5];  // [4][256]
    const float* fc1_w  = (const float*)d_in[6];  // [512][256]
    const float* fc1_b  = (const float*)d_in[7];
    const float* fc2_w  = (const float*)d_in[8];  // [256][16]
    const float* fc2_b  = (const float*)d_in[9];
    const int* src = eidx;
    const int* dst = eidx + N_EDGES;

    const size_t ND = (size_t)N_NODES * D;
    float* ws = (float*)d_ws;
    size_t off = 0;
    float* HR     = ws + off; off += ND;
    float* BASE   = ws + off; off += ND;
    float* AGG    = ws + off; off += ND;        // CNT directly after AGG so a
    float* CNT    = ws + off; off += N_NODES;   // single zero pass clears both
    float* HTMP   = ws + off; off += ND;
    float* EMB0   = ws + off; off += ND;
    float* EMB1   = ws + off; off += ND;
    float* LOGITS = ws + off; off += (size_t)N_NODES * LL_OUT;
    (void)ws_size; (void)in_sizes; (void)n_in; (void)out_size;

    const dim3 blk(256);
    auto gemm = [&](const float* A, const float* B, const float* bv, float* C,
                    int Dout, int flags) {
        dim3 grid((Dout + 31) / 32, (N_NODES + 127) / 128);
        wmma_gemm_f32<<<grid, blk, 0, stream>>>(A, B, bv, C, N_NODES, Dout, flags);
    };

    auto conv = [&](int ci, int rel, const float* hin, float* hout) {
        const size_t nz = ND + N_NODES;
        zero_f32<<<2048, blk, 0, stream>>>(AGG, nz);
        gemm(hin, rel_W + ((size_t)ci * NUM_REL + rel) * D * D, nullptr, HR, D, 0);
        gemm(hin, root_W + (size_t)ci * D * D, biasv + ci * D, BASE, D, GF_BIAS);
        edge_scatter<<<(N_EDGES + 3) / 4, blk, 0, stream>>>(HR, src, dst, etype,
                                                            rel, AGG, CNT);
        conv_finalize<<<(unsigned)((ND + 255) / 256), blk, 0, stream>>>(AGG, CNT,
                                                                        BASE, hout);
    };

    // METAPATHS = [[0,1],[2,3]] -> conv index ci = 0..3, relation = ci
    conv(0, 0, x,    HTMP);
    conv(1, 1, HTMP, EMB0);
    conv(2, 2, x,    HTMP);
    conv(3, 3, HTMP, EMB1);

    // fc1: hcat @ fc1_w == EMB0 @ fc1_w[0:256] + EMB1 @ fc1_w[256:512]
    gemm(EMB0, fc1_w,                 fc1_b,   HR, D, GF_BIAS);
    gemm(EMB1, fc1_w + (size_t)D * D, nullptr, HR, D, GF_ACC | GF_RELU);

    // fc2 -> logits (Dout = 16: one column tile, LDS half-panel zero-filled)
    gemm(HR, fc2_w, fc2_b, LOGITS, LL_OUT, GF_BIAS);

    log_softmax16<<<(N_NODES + 255) / 256, blk, 0, stream>>>(LOGITS, (float*)d_out);
}